// CumulativeIFFT_50294067036690
// MI455X (gfx1250) — compile-verified
//
#include <hip/hip_runtime.h>
#include <math.h>

typedef __attribute__((ext_vector_type(2))) float v2f;
typedef __attribute__((ext_vector_type(8))) float v8f;

#define SEQ    512
#define DSP    64
#define KMAX   32
#define KK     64      // 2*KMAX (cos block | sin block)
#define NSLICE 128     // n rows per block
#define TSTR   66      // twiddle LDS row stride (floats), row = local n
#define SSTR   66      // spectral LDS row stride (floats), row = d, cols = kk

__global__ __launch_bounds__(128)
void cifft_wmma_f32(const float* __restrict__ spectral, float* __restrict__ out)
{
    __shared__ float Tl[NSLICE * TSTR]; // twiddle slice: row r (= local n), cols [cos(0..31) | sin(0..31)]
    __shared__ float Sl[DSP * SSTR];    // scaled spectral, TRANSPOSED: row d, col kk

    const int tid = threadIdx.x;
    const int bt  = blockIdx.x >> 2;   // 0..2047  (b*512 + t)
    const int nsl = blockIdx.x & 3;    // which 128-row n slice
    const int t   = bt & (SEQ - 1);
    const float scale = sqrtf((float)(t + 1)) * (1.0f / (float)SEQ);

    // ---- Phase 0: load spectral[bt] (64 d x 32 k x 2), scale, fold signs, into S[d][kk] ----
    {
        const float4* src = (const float4*)(spectral + (size_t)bt * (DSP * KMAX * 2));
        #pragma unroll
        for (int i = 0; i < 8; ++i) {
            int f = i * 128 + tid;            // float4 index 0..1023
            float4 v = src[f];
            int d = f >> 4;                   // f/16
            int k = (f & 15) * 2;             // even k; float4 = (re_k, im_k, re_k+1, im_k+1)
            float2 re; re.x =  v.x * scale; re.y =  v.z * scale;   // cols k, k+1
            float2 im; im.x = -v.y * scale; im.y = -v.w * scale;   // cols 32+k, 33+k
            *(float2*)&Sl[d * SSTR + k]        = re;
            *(float2*)&Sl[d * SSTR + KMAX + k] = im;
        }
    }

    // ---- Phase 1: twiddle slice. Thread tid owns row r = tid -> n = nsl*128 + r ----
    {
        const int r = tid;
        const int n = nsl * NSLICE + r;
        const float c0 = 6.2831853071795864769f / (float)SEQ;
        #pragma unroll 4
        for (int k = 0; k < KMAX; ++k) {
            int mm = (n * k) & (SEQ - 1);     // exact period reduction: angle in [0, 2pi)
            float sv, cv;
            __sincosf(c0 * (float)mm, &sv, &cv);
            Tl[r * TSTR + k]        = cv;
            Tl[r * TSTR + KMAX + k] = sv;
        }
    }

    __syncthreads();

    // ---- Phase 2: per-wave GEMM, 32 n-rows x 64 d, K = 64 in steps of 4 ----
    const int wave  = tid >> 5;
    const int lane  = tid & 31;
    const int h     = lane >> 4;   // half-wave index
    const int m     = lane & 15;
    const int rbase = wave * 32;   // this wave's first local n row

    v8f acc[2][4] = {};            // [row tile][d tile], zero-initialized

    #pragma unroll 4
    for (int k0 = 0; k0 < KK; k0 += 4) {
        v2f A[2], Bm[4];
        #pragma unroll
        for (int rt = 0; rt < 2; ++rt) {
            // A lane L: T[n0 + L%16, k0 + 2*(L/16) + {0,1}]  -> one 8B-aligned b64 LDS load
            A[rt] = *(const v2f*)&Tl[(rbase + rt * 16 + m) * TSTR + k0 + 2 * h];
        }
        #pragma unroll
        for (int c = 0; c < 4; ++c) {
            // B VGPR j lane L: S[k0 + 2*(L/16) + j, d0 + L%16]; transposed layout
            // makes the (j=0, j=1) pair contiguous -> one b64 LDS load, no repack movs
            Bm[c] = *(const v2f*)&Sl[(c * 16 + m) * SSTR + k0 + 2 * h];
        }
        #pragma unroll
        for (int rt = 0; rt < 2; ++rt)
            #pragma unroll
            for (int c = 0; c < 4; ++c)
                acc[rt][c] = __builtin_amdgcn_wmma_f32_16x16x4_f32(
                    false, A[rt], false, Bm[c], (short)0, acc[rt][c], false, false);
    }

    // ---- Phase 3: stage this wave's 32x64 tile into its OWN (consumed) twiddle rows,
    //      then fully-coalesced float4 global stores. Wave-private => no barrier. ----
    #pragma unroll
    for (int rt = 0; rt < 2; ++rt)
        #pragma unroll
        for (int c = 0; c < 4; ++c)
            #pragma unroll
            for (int v = 0; v < 8; ++v)
                // D lane L VGPR v: row = v + 8*(L/16), col = d0 + L%16
                Tl[(rbase + rt * 16 + v + 8 * h) * TSTR + c * 16 + m] = acc[rt][c][v];

    const size_t gbase = ((size_t)bt * SEQ + (size_t)(nsl * NSLICE + rbase)) * DSP;
    #pragma unroll
    for (int i = 0; i < 16; ++i) {
        int f   = i * 128 + lane * 4;   // 2048 floats per wave tile
        int row = f >> 6;
        int col = f & 63;
        const float* p = &Tl[(rbase + row) * TSTR + col];  // 8B-aligned
        float2 lo = *(const float2*)p;
        float2 hi = *(const float2*)(p + 2);
        float4 o; o.x = lo.x; o.y = lo.y; o.z = hi.x; o.w = hi.y;
        *(float4*)(out + gbase + f) = o;                   // 16B-aligned, 256B per half-wave
    }
}

extern "C" void kernel_launch(void* const* d_in, const int* in_sizes, int n_in,
                              void* d_out, int out_size, void* d_ws, size_t ws_size,
                              hipStream_t stream) {
    (void)in_sizes; (void)n_in; (void)out_size; (void)d_ws; (void)ws_size;
    const float* spectral = (const float*)d_in[0];
    float* out = (float*)d_out;
    // grid = B*T * (SEQ/NSLICE) = 2048 * 4
    cifft_wmma_f32<<<dim3(8192), dim3(128), 0, stream>>>(spectral, out);
}